// GNNClassifier_52913997087420
// MI455X (gfx1250) — compile-verified
//
#include <hip/hip_runtime.h>
#include <math.h>

// GAT 2-layer forward for MI455X (gfx1250).
// GEMMs use V_WMMA_F32_16X16X4_F32 (fp32 WMMA, full precision vs reference).
// Each wave computes a 16x64 output tile (4 accumulators) so the A fragment is
// reused across 4 WMMAs per k-step -> WMMA-issue bound instead of load bound.
// Edge softmax/aggregation uses fp32 global atomics; feature arrays (~102MB)
// fit in the 192MB L2 so the random gather/scatter stays on-die.
// Workspace requirement: ~135.4 MB.

#define NNODES 50000
#define NEDGES 800000
#define IN_DIM 256
#define HID 64
#define HEADS 4
#define F1 (HEADS * HID)   // 256
#define OUT_DIM 64
#define NEG_SLOPE 0.2f

typedef float v2f __attribute__((ext_vector_type(2)));
typedef float v8f __attribute__((ext_vector_type(8)));

// ---------------------------------------------------------------------------
// fp32 WMMA GEMM: C[M,N] = A[M,K] @ B[K,N], row-major, 16|M, (16*NT)|N, 4|K.
// One wave computes a 16 x (16*NT) tile: one A fragment feeds NT WMMAs per
// k-step. blockDim = (32, 8) -> 8 wave-tiles per block. All branches are
// wave-uniform so EXEC is all-1s for every WMMA.
// ---------------------------------------------------------------------------
template <int NT>
__global__ void gemm_wmma_f32(const float* __restrict__ A,
                              const float* __restrict__ B,
                              float* __restrict__ C,
                              int M, int N, int K) {
    const int lane  = threadIdx.x;                              // 0..31
    const int wtile = blockIdx.x * blockDim.y + threadIdx.y;    // wave-uniform
    const int ngroups = N / (16 * NT);
    const int total   = (M >> 4) * ngroups;
    if (wtile >= total) return;                                 // uniform per wave

    const int tm   = wtile / ngroups;
    const int tg   = wtile - tm * ngroups;
    const int row0 = tm << 4;
    const int col0 = tg * (16 * NT);

    const int mloc  = lane & 15;            // matrix row (A) / col (B,D)
    const int khalf = (lane >> 4) << 1;     // k-pair base: 0 or 2

    const float* arow  = A + (size_t)(row0 + mloc) * K;
    const float* bbase = B + col0 + mloc;

    v8f acc[NT];
    const v8f zero = {};
#pragma unroll
    for (int t = 0; t < NT; ++t) acc[t] = zero;

#pragma unroll 4
    for (int k0 = 0; k0 < K; k0 += 4) {
        const int k = k0 + khalf;
        v2f a;
        a.x = arow[k];
        a.y = arow[k + 1];
        const float* b0 = bbase + (size_t)k * N;
        const float* b1 = bbase + (size_t)(k + 1) * N;
#pragma unroll
        for (int t = 0; t < NT; ++t) {
            v2f b;
            b.x = b0[t * 16];
            b.y = b1[t * 16];
            // D = A(16x4) * B(4x16) + C  -> v_wmma_f32_16x16x4_f32
            acc[t] = __builtin_amdgcn_wmma_f32_16x16x4_f32(
                /*neg_a=*/false, a, /*neg_b=*/false, b,
                /*c_mod=*/(short)0, acc[t], /*reuse_a=*/false, /*reuse_b=*/false);
        }
    }

    // D layout: VGPR v -> row (v + 8*(lane>=16)), col = lane&15
    const int rbase = row0 + ((lane >> 4) << 3);
#pragma unroll
    for (int t = 0; t < NT; ++t) {
        const int c = col0 + t * 16 + mloc;
#pragma unroll
        for (int v = 0; v < 8; ++v) {
            C[(size_t)(rbase + v) * N + c] = acc[t][v];
        }
    }
}

// ---------------------------------------------------------------------------
// Init: zero accumulators, set running-max buffers to -inf. Re-run every call.
// ---------------------------------------------------------------------------
__global__ void init_kernel(float* out1, float* out2, float* s1, float* s2,
                            float* m1, float* m2) {
    const int tid = blockIdx.x * blockDim.x + threadIdx.x;
    const int stride = gridDim.x * blockDim.x;
    for (int i = tid; i < NNODES * F1; i += stride) out1[i] = 0.0f;
    for (int i = tid; i < NNODES * OUT_DIM; i += stride) out2[i] = 0.0f;
    for (int i = tid; i < NNODES * HEADS; i += stride) { s1[i] = 0.0f; m1[i] = -__builtin_inff(); }
    for (int i = tid; i < NNODES; i += stride) { s2[i] = 0.0f; m2[i] = -__builtin_inff(); }
}

// el[n,h] = <feat[n,h,:], al[h,:]> ; er likewise. Layer 1 (H=4, D=64).
__global__ void eler1_kernel(const float* __restrict__ feat,
                             const float* __restrict__ al,
                             const float* __restrict__ ar,
                             float* __restrict__ el, float* __restrict__ er) {
    const int i = blockIdx.x * blockDim.x + threadIdx.x;
    if (i >= NNODES * HEADS) return;
    const int n = i / HEADS, h = i - n * HEADS;
    const float* f = feat + (size_t)n * F1 + h * HID;
    const float* a = al + h * HID;
    const float* b = ar + h * HID;
    float sl = 0.0f, sr = 0.0f;
    for (int d = 0; d < HID; ++d) { const float v = f[d]; sl += v * a[d]; sr += v * b[d]; }
    el[i] = sl; er[i] = sr;
}

// Layer 2 (H=1, D=64).
__global__ void eler2_kernel(const float* __restrict__ feat,
                             const float* __restrict__ al,
                             const float* __restrict__ ar,
                             float* __restrict__ el, float* __restrict__ er) {
    const int n = blockIdx.x * blockDim.x + threadIdx.x;
    if (n >= NNODES) return;
    const float* f = feat + (size_t)n * OUT_DIM;
    float sl = 0.0f, sr = 0.0f;
    for (int d = 0; d < OUT_DIM; ++d) { const float v = f[d]; sl += v * al[d]; sr += v * ar[d]; }
    el[n] = sl; er[n] = sr;
}

// Float atomic-max via the sign-split int/uint trick (valid for mixed signs).
__device__ __forceinline__ void atomic_max_f32(float* addr, float val) {
    if (val >= 0.0f) atomicMax((int*)addr, __float_as_int(val));
    else             atomicMin((unsigned int*)addr, __float_as_uint(val));
}

// Pass 1: e = leaky_relu(el[src]+er[dst]); running segment max over dst.
__global__ void edge_max_kernel(const int* __restrict__ src, const int* __restrict__ dst,
                                const float* __restrict__ el, const float* __restrict__ er,
                                float* __restrict__ e, float* __restrict__ m, int H) {
    const int i = blockIdx.x * blockDim.x + threadIdx.x;
    if (i >= NEDGES * H) return;
    const int eid = i / H, h = i - eid * H;
    float v = el[src[eid] * H + h] + er[dst[eid] * H + h];
    v = v > 0.0f ? v : NEG_SLOPE * v;
    e[i] = v;
    atomic_max_f32(&m[dst[eid] * H + h], v);
}

// Pass 2: segment sum of exp(e - m[dst]).
__global__ void edge_sum_kernel(const int* __restrict__ dst,
                                const float* __restrict__ e, const float* __restrict__ m,
                                float* __restrict__ s, int H) {
    const int i = blockIdx.x * blockDim.x + threadIdx.x;
    if (i >= NEDGES * H) return;
    const int eid = i / H, h = i - eid * H;
    const float w = __expf(e[i] - m[dst[eid] * H + h]);
    atomicAdd(&s[dst[eid] * H + h], w);
}

// Pass 3 layer 1: out[dst] += feat[src] * alpha. One block (256 thr) per edge.
__global__ void edge_agg1_kernel(const int* __restrict__ src, const int* __restrict__ dst,
                                 const float* __restrict__ feat,
                                 const float* __restrict__ e, const float* __restrict__ m,
                                 const float* __restrict__ s, float* __restrict__ out) {
    const int eid = blockIdx.x;
    const int t = threadIdx.x;               // 0..255 -> (h = t>>6, d = t&63)
    const int h = t >> 6;
    const int nd = dst[eid];
    const int si = nd * HEADS + h;
    const float alpha = __expf(e[eid * HEADS + h] - m[si]) / s[si];
    atomicAdd(&out[(size_t)nd * F1 + t], feat[(size_t)src[eid] * F1 + t] * alpha);
}

// Pass 3 layer 2 (H=1): one block (64 thr) per edge, accumulate into d_out.
__global__ void edge_agg2_kernel(const int* __restrict__ src, const int* __restrict__ dst,
                                 const float* __restrict__ feat,
                                 const float* __restrict__ e, const float* __restrict__ m,
                                 const float* __restrict__ s, float* __restrict__ out) {
    const int eid = blockIdx.x;
    const int d = threadIdx.x;               // 0..63
    const int nd = dst[eid];
    const float alpha = __expf(e[eid] - m[nd]) / s[nd];
    atomicAdd(&out[(size_t)nd * OUT_DIM + d], feat[(size_t)src[eid] * OUT_DIM + d] * alpha);
}

// ELU in place: v > 0 ? v : exp(v) - 1.
__global__ void elu_kernel(float* __restrict__ a) {
    const int i = blockIdx.x * blockDim.x + threadIdx.x;
    if (i >= NNODES * F1) return;
    const float v = a[i];
    a[i] = v > 0.0f ? v : (__expf(v) - 1.0f);
}

// ---------------------------------------------------------------------------
extern "C" void kernel_launch(void* const* d_in, const int* in_sizes, int n_in,
                              void* d_out, int out_size, void* d_ws, size_t ws_size,
                              hipStream_t stream) {
    const float* x   = (const float*)d_in[0];
    const int*   src = (const int*)  d_in[1];
    const int*   dst = (const int*)  d_in[2];
    const float* W1  = (const float*)d_in[3];
    const float* al1 = (const float*)d_in[4];
    const float* ar1 = (const float*)d_in[5];
    const float* W2  = (const float*)d_in[6];
    const float* al2 = (const float*)d_in[7];
    const float* ar2 = (const float*)d_in[8];
    float* out = (float*)d_out;

    // Workspace layout (floats). Total ~33.85M floats = ~135.4 MB.
    float* w = (float*)d_ws;
    float* feat1 = w; w += (size_t)NNODES * F1;        // 12.8M
    float* out1  = w; w += (size_t)NNODES * F1;        // 12.8M (becomes h1 after ELU)
    float* el1   = w; w += (size_t)NNODES * HEADS;
    float* er1   = w; w += (size_t)NNODES * HEADS;
    float* m1    = w; w += (size_t)NNODES * HEADS;
    float* s1    = w; w += (size_t)NNODES * HEADS;
    float* e1    = w; w += (size_t)NEDGES * HEADS;     // 3.2M
    float* feat2 = w; w += (size_t)NNODES * OUT_DIM;   // 3.2M
    float* el2   = w; w += (size_t)NNODES;
    float* er2   = w; w += (size_t)NNODES;
    float* m2    = w; w += (size_t)NNODES;
    float* s2    = w; w += (size_t)NNODES;
    float* e2    = w; w += (size_t)NEDGES;             // 0.8M

    init_kernel<<<2048, 256, 0, stream>>>(out1, out, s1, s2, m1, m2);

    // ---- Layer 1 ----
    {   // feat1 = x @ W1 : M=50000, N=256, K=256
        // 16x64 per wave -> 3125 * 4 = 12500 wave-tiles, 8 waves/block
        const int wtiles = (NNODES / 16) * (F1 / 64);
        dim3 blk(32, 8);
        gemm_wmma_f32<4><<<(wtiles + 7) / 8, blk, 0, stream>>>(x, W1, feat1, NNODES, F1, IN_DIM);
    }
    eler1_kernel<<<(NNODES * HEADS + 255) / 256, 256, 0, stream>>>(feat1, al1, ar1, el1, er1);
    edge_max_kernel<<<(NEDGES * HEADS + 255) / 256, 256, 0, stream>>>(src, dst, el1, er1, e1, m1, HEADS);
    edge_sum_kernel<<<(NEDGES * HEADS + 255) / 256, 256, 0, stream>>>(dst, e1, m1, s1, HEADS);
    edge_agg1_kernel<<<NEDGES, 256, 0, stream>>>(src, dst, feat1, e1, m1, s1, out1);
    elu_kernel<<<(NNODES * F1 + 255) / 256, 256, 0, stream>>>(out1);

    // ---- Layer 2 ----
    {   // feat2 = h1 @ W2 : M=50000, N=64, K=256 -> 3125 wave-tiles
        const int wtiles = (NNODES / 16) * (OUT_DIM / 64);
        dim3 blk(32, 8);
        gemm_wmma_f32<4><<<(wtiles + 7) / 8, blk, 0, stream>>>(out1, W2, feat2, NNODES, OUT_DIM, F1);
    }
    eler2_kernel<<<(NNODES + 255) / 256, 256, 0, stream>>>(feat2, al2, ar2, el2, er2);
    edge_max_kernel<<<(NEDGES + 255) / 256, 256, 0, stream>>>(src, dst, el2, er2, e2, m2, 1);
    edge_sum_kernel<<<(NEDGES + 255) / 256, 256, 0, stream>>>(dst, e2, m2, s2, 1);
    edge_agg2_kernel<<<NEDGES, 64, 0, stream>>>(src, dst, feat2, e2, m2, s2, out);
    // H=1 -> mean over heads is identity; d_out holds the logits.
}